// PNAConv_936302871070
// MI455X (gfx1250) — compile-verified
//
#include <hip/hip_runtime.h>
#include <math.h>

typedef __attribute__((ext_vector_type(2))) float v2f;
typedef __attribute__((ext_vector_type(8))) float v8f;
typedef __attribute__((ext_vector_type(4))) unsigned int u32x4;
typedef __attribute__((ext_vector_type(4))) int i32x4;
typedef __attribute__((ext_vector_type(8))) int i32x8;

#define CH 128          // IN_CH == OUT_CH == 128
#define LSTR 132        // LDS row stride (floats): 4-DWORD skew -> conflict-free 16x4 frag loads
#define NT 64           // node rows per GEMM block tile

// float atomic max via monotone integer bit trick (init value must be -inf)
__device__ __forceinline__ void atomic_max_f32(float* addr, float v) {
  if (v >= 0.0f) {
    atomicMax((int*)addr, __float_as_int(v));
  } else {
    atomicMin((unsigned int*)addr, __float_as_uint(v));
  }
}

// ---------------- TDM: async load of a [tile_d1 x 128] f32 tile into skewed LDS ----------------
// D# per CDNA5 ISA. Padding: after every 128 DWORDs insert 4 DWORDs
// -> LDS row stride 132 floats, matching LSTR. tensor_dim1 = remaining rows gives OOB zero-fill.
__device__ __forceinline__ void tdm_load_tile_f32(unsigned lds_addr, const float* gbase,
                                                  unsigned rows_left, unsigned tile_d1) {
  const unsigned long long ga = (unsigned long long)(uintptr_t)gbase;
  u32x4 g0;
  g0[0] = 1u;                                               // count=1, user D#
  g0[1] = lds_addr;                                         // LDS byte address
  g0[2] = (unsigned)(ga & 0xFFFFFFFFull);                   // global_addr[31:0]
  g0[3] = (unsigned)((ga >> 32) & 0x1FFFFFFull)             // global_addr[56:32]
        | (2u << 30);                                       // type=2 ("image")

  union { unsigned long long q[4]; i32x8 v; } g1;
  g1.q[0] = (2ull << 16)                                    // data_size = 4 bytes
          | (1ull << 20)                                    // pad_enable
          | (6ull << 22)                                    // pad_interval: 128 DWORDs
          | (3ull << 25)                                    // pad_amount: 4 DWORDs
          | ((unsigned long long)(CH & 0xFFFF) << 48);      // tensor_dim0[15:0] = 128
  g1.q[1] = 0ull                                            // tensor_dim0[31:16]
          | ((unsigned long long)rows_left << 16)           // tensor_dim1 (OOB rows -> 0)
          | ((unsigned long long)CH << 48);                 // tile_dim0 = 128
  g1.q[2] = (unsigned long long)tile_d1                     // tile_dim1 = rows in tile
          | ((unsigned long long)CH << 32);                 // tensor_dim0_stride = 128
  g1.q[3] = 0ull;                                          // tensor_dim1_stride unused (2D)

  const i32x4 gz = {0, 0, 0, 0};
#if __clang_major__ >= 23
  const i32x8 gz8 = {0, 0, 0, 0, 0, 0, 0, 0};
  __builtin_amdgcn_tensor_load_to_lds(g0, g1.v, gz, gz, gz8, 0);
#else
  __builtin_amdgcn_tensor_load_to_lds(g0, g1.v, gz, gz, 0);
#endif
}

// ---------------- stage 0: workspace init ----------------
__global__ __launch_bounds__(256) void pna_init(float* __restrict__ degree,
                                                float* __restrict__ scal,
                                                float* __restrict__ aggs,
                                                float* __restrict__ aggm,
                                                int n) {
  const int stride = gridDim.x * blockDim.x;
  const int i0 = blockIdx.x * blockDim.x + threadIdx.x;
  const int total = n * CH;
  for (int i = i0; i < total; i += stride) {
    aggs[i] = 0.0f;
    aggm[i] = -INFINITY;
  }
  for (int i = i0; i < n; i += stride) degree[i] = 0.0f;
  if (i0 == 0) { scal[0] = 0.0f; scal[1] = 1.0f; }
}

// ---------------- stage 1: edge scatter (L2-resident atomics) ----------------
__global__ __launch_bounds__(128) void pna_scatter(const float* __restrict__ x,
                                                   const long long* __restrict__ ei,
                                                   float* __restrict__ degree,
                                                   float* __restrict__ aggs,
                                                   float* __restrict__ aggm,
                                                   int nEdges) {
  const int e = blockIdx.x;
  if (e >= nEdges) return;
  const int t = threadIdx.x;                          // channel
  const size_t src = (size_t)ei[e];                   // edge_index[0][e]
  const size_t dst = (size_t)ei[(size_t)nEdges + e];  // edge_index[1][e]
  const float v = x[src * CH + t];
  atomicAdd(&aggs[dst * CH + t], v);
  atomic_max_f32(&aggm[dst * CH + t], v);
  if (t == 0) atomicAdd(&degree[dst], 1.0f);
}

// ---------------- stage 2: mean(log1p(deg+1)) reduction ----------------
__global__ __launch_bounds__(256) void pna_degree_reduce(const float* __restrict__ degree,
                                                         float* __restrict__ scal,
                                                         int n) {
  __shared__ float sred[256];
  const int i = blockIdx.x * 256 + threadIdx.x;
  float v = 0.0f;
  if (i < n) v = log1pf(degree[i] + 1.0f);
  sred[threadIdx.x] = v;
  __syncthreads();
  for (int s = 128; s > 0; s >>= 1) {
    if (threadIdx.x < s) sred[threadIdx.x] += sred[threadIdx.x + s];
    __syncthreads();
  }
  if (threadIdx.x == 0) atomicAdd(&scal[0], sred[0]);
}

__global__ void pna_finalize(float* scal, int n) {
  scal[1] = fmaxf(scal[0] / (float)n, 1.0f);          // degree_reference
}

// ---------------- stage 3: fused PNA GEMM via V_WMMA_F32_16X16X4_F32 ----------------
// Block: 256 threads = 8 wave32. Block tile: 64 nodes x 128 out channels.
// Wave w owns out channels [16w,16w+16) and 4 node sub-tiles -> one B fragment feeds 4 WMMAs.
// Features staged by the Tensor Data Mover into skew-padded LDS (stride 132 floats).
__global__ __launch_bounds__(256) void pna_gemm_wmma(const float* __restrict__ x,
                                                     const float* __restrict__ Wmsg,
                                                     const float* __restrict__ bmsg,
                                                     const float* __restrict__ Wroot,
                                                     const float* __restrict__ broot,
                                                     const float* __restrict__ degree,
                                                     const float* __restrict__ aggs,
                                                     const float* __restrict__ aggm,
                                                     const float* __restrict__ scal,
                                                     float* __restrict__ out,
                                                     int n) {
  __shared__ float sX[NT * LSTR];
  __shared__ float sS[NT * LSTR];
  __shared__ float sM[NT * LSTR];
  __shared__ float sDeg[NT];

  const int tid = threadIdx.x;
  const int nodeBase = blockIdx.x * NT;
  const unsigned rowsLeft = (unsigned)(n - nodeBase);   // grid guarantees > 0

  // Wave 0 issues the three async TDM tile loads; TDM ignores EXEC, 1 issue per wave.
  if (tid < 32) {
    tdm_load_tile_f32((unsigned)(uintptr_t)&sX[0], x    + (size_t)nodeBase * CH, rowsLeft, NT);
    tdm_load_tile_f32((unsigned)(uintptr_t)&sS[0], aggs + (size_t)nodeBase * CH, rowsLeft, NT);
    tdm_load_tile_f32((unsigned)(uintptr_t)&sM[0], aggm + (size_t)nodeBase * CH, rowsLeft, NT);
  }

  for (int r = tid; r < NT; r += 256) {
    const int nn = nodeBase + r;
    sDeg[r] = (nn < n) ? degree[nn] : 0.0f;
  }

  if (tid < 32) __builtin_amdgcn_s_wait_tensorcnt(0);   // TDM done before barrier
  __syncthreads();                                      // publish LDS to all 8 waves

  // Empty segments: segment_max is -inf in ws; reference replaces with 0 when degree==0.
  for (int idx = tid; idx < NT * CH; idx += 256) {
    const int r = idx >> 7;
    if (!(sDeg[r] > 0.0f)) sM[r * LSTR + (idx & (CH - 1))] = 0.0f;
  }
  __syncthreads();

  const int lane = tid & 31;
  const int wave = tid >> 5;
  const int row = lane & 15;            // A row (within sub-tile) / B col / C col
  const int khalf = (lane >> 4) << 1;   // K offset 0 (lanes 0-15) or 2 (lanes 16-31)
  const int o = wave * 16 + row;        // global output channel for this lane's B/C column

  // Per-node PNA scalers for the 4 node sub-tiles handled by this lane's rows.
  const float dref = scal[1];
  float amp[4], att[4], invd[4];
#pragma unroll
  for (int mt = 0; mt < 4; ++mt) {
    const float deg = sDeg[mt * 16 + row];
    const float dt = log1pf(deg + 1.0f);
    amp[mt]  = dt / dref;
    att[mt]  = dref / fmaxf(dt, 1e-6f);
    invd[mt] = 1.0f / fmaxf(deg, 1.0f);
  }

  v8f acc[4] = {{}, {}, {}, {}};

  for (int j = 0; j < 10; ++j) {
    // feature: j<6 -> agg_sum (mean = sum*invd), 6..8 -> agg_max, 9 -> x (root)
    const float* fbase = ((j < 6) ? sS : (j < 9) ? sM : sX) + row * LSTR + khalf;
    const float* wf = (j < 9) ? (Wmsg + (size_t)o * (9 * CH) + j * CH + khalf)
                              : (Wroot + (size_t)o * CH + khalf);
    const int jm = j - (j / 3) * 3;
    float sj[4];
#pragma unroll
    for (int mt = 0; mt < 4; ++mt) {
      float s = (jm == 1) ? amp[mt] : (jm == 2) ? att[mt] : 1.0f;
      if (j >= 3 && j < 6) s *= invd[mt];
      if (j == 9) s = 1.0f;
      sj[mt] = s;
    }

#pragma unroll 8
    for (int k = 0; k < CH; k += 4) {
      const float2 bvv = *(const float2*)(wf + k);      // one B frag feeds 4 WMMAs
      v2f b; b[0] = bvv.x; b[1] = bvv.y;
#pragma unroll
      for (int mt = 0; mt < 4; ++mt) {
        const float2 avv = *(const float2*)(fbase + mt * (16 * LSTR) + k);
        v2f a; a[0] = avv.x * sj[mt]; a[1] = avv.y * sj[mt];
        acc[mt] = __builtin_amdgcn_wmma_f32_16x16x4_f32(
            /*neg_a=*/false, a, /*neg_b=*/false, b,
            /*c_mod=*/(short)0, acc[mt], /*reuse_a=*/false, /*reuse_b=*/false);
      }
    }
  }

  // C/D layout: VGPR r holds row (r + 8*(lane/16)), column lane%16.
  const float bias = bmsg[o] + broot[o];
#pragma unroll
  for (int mt = 0; mt < 4; ++mt) {
    const int mBase = nodeBase + mt * 16 + ((lane >> 4) << 3);
#pragma unroll
    for (int r = 0; r < 8; ++r) {
      const int m = mBase + r;
      if (m < n) out[(size_t)m * CH + o] = acc[mt][r] + bias;
    }
  }
}

extern "C" void kernel_launch(void* const* d_in, const int* in_sizes, int n_in,
                              void* d_out, int out_size, void* d_ws, size_t ws_size,
                              hipStream_t stream) {
  const float*     x     = (const float*)d_in[0];
  const long long* ei    = (const long long*)d_in[1];   // int64 edge_index [2,E]
  const float*     Wmsg  = (const float*)d_in[2];       // [128, 1152]
  const float*     bmsg  = (const float*)d_in[3];
  const float*     Wroot = (const float*)d_in[4];       // [128, 128]
  const float*     broot = (const float*)d_in[5];
  float*           out   = (float*)d_out;

  const int n  = in_sizes[0] / CH;   // 50000 nodes
  const int nE = in_sizes[1] / 2;    // 600000 edges

  // Workspace layout: degree[n] | scal[64] | agg_sum[n*128] | agg_max[n*128]
  float* degree = (float*)d_ws;
  float* scal   = degree + n;
  float* aggs   = scal + 64;
  float* aggm   = aggs + (size_t)n * CH;

  pna_init<<<2048, 256, 0, stream>>>(degree, scal, aggs, aggm, n);
  pna_scatter<<<nE, 128, 0, stream>>>(x, ei, degree, aggs, aggm, nE);
  pna_degree_reduce<<<(n + 255) / 256, 256, 0, stream>>>(degree, scal, n);
  pna_finalize<<<1, 1, 0, stream>>>(scal, n);
  pna_gemm_wmma<<<(n + NT - 1) / NT, 256, 0, stream>>>(x, Wmsg, bmsg, Wroot, broot,
                                                       degree, aggs, aggm, scal, out, n);
}